// ComplexEMA_1503238554305
// MI455X (gfx1250) — compile-verified
//
#include <hip/hip_runtime.h>
#include <math.h>
#include <stdint.h>

// ComplexEMA for MI455X (gfx1250): chunked linear-scan + fp32 WMMA Toeplitz pass.
// B=8, D=1024, L=2048, N=16 poles, chunk C=64, 32 chunks.

typedef float v2f __attribute__((ext_vector_type(2)));
typedef float v8f __attribute__((ext_vector_type(8)));

#define NPOLE 16
#define CHUNK 64
#define NCH   32
#define SEQL  2048

__device__ __forceinline__ float sigmoidf_(float v) { return 1.0f / (1.0f + expf(-v)); }

// CDNA5 async global->LDS copy, 8 bytes per lane (ASYNCcnt-tracked).
__device__ __forceinline__ void async_ld_b64(uint32_t lds_off, const float* gptr) {
    asm volatile("global_load_async_to_lds_b64 %0, %1, off"
                 :: "v"(lds_off), "v"(gptr) : "memory");
}

__global__ __launch_bounds__(256)
void cema_scan_wmma_kernel(const float* __restrict__ x,
                           const float* __restrict__ alpha,
                           const float* __restrict__ delta,
                           const float* __restrict__ theta,
                           const float* __restrict__ gre,
                           const float* __restrict__ gim,
                           const float* __restrict__ omega,
                           float* __restrict__ out,
                           int Bn, int D)
{
    const int d    = blockIdx.x;
    const int tid  = threadIdx.x;
    const int lane = tid & 31;
    const int wv   = tid >> 5;      // wave id 0..7
    const int hn   = lane & 15;     // pole / N-column / M-row within fragment
    const int half = lane >> 4;     // fragment half (K-offset +2, M-offset +8)

    __shared__ float sPhi[NPOLE], sL2r[NPOLE], sGr[NPOLE], sGi[NPOLE], sQCr[NPOLE], sQCi[NPOLE];
    __shared__ float sW[NPOLE * CHUNK * 2];   // w[n][u] = q^(63-u), interleaved re/im (8KB)
    __shared__ float sEE[CHUNK * 32];         // [t][0..15]=Re(gp q^{t+1}); [t][16..31]=-Im(gp q^{t+1}) (8KB)
    __shared__ float sTapR[128];              // zero-padded reversed taps: [j<64]=k[63-j], [j>=64]=0
    __shared__ float sSS[8 * NCH * 32];       // states per (b,chunk): [Sre(16)|Sim(16)] (32KB)
    __shared__ float sXA[8][2][CHUNK];        // per-wave async x chunk double buffer (4KB)

    // ---- stage 1: per-pole scalars ----
    if (tid < NPOLE) {
        const int n = tid;
        const int gidx = d * NPOLE + n;
        const float a   = sigmoidf_(alpha[gidx]);
        const float dd  = sigmoidf_(delta[gidx]);
        const float th  = sigmoidf_(theta[d]) * (6.28318530717958647692f / (float)NPOLE);
        const float phi = (float)(n + 1) * th;
        const float radius = fminf(1.0f - a * dd, 1.0f);   // in (0,1)
        const float l2r = log2f(radius);
        sPhi[n] = phi; sL2r[n] = l2r;
        sGr[n]  = 0.25f * gre[gidx] * a;                   // scale = 1/sqrt(16)
        sGi[n]  = 0.25f * gim[gidx] * a;
        const float rC = exp2f((float)CHUNK * l2r);        // radius^64
        const float aC = (float)CHUNK * phi;
        sQCr[n] = rC * cosf(aC);
        sQCi[n] = rC * sinf(aC);
    }
    __syncthreads();

    // ---- stage 2: tables ----
    for (int idx = tid; idx < NPOLE * CHUNK; idx += 256) {   // w[n][u] = q^(63-u)
        const int n = idx >> 6, u = idx & 63;
        const float e  = (float)(CHUNK - 1 - u);
        const float rr = exp2f(e * sL2r[n]);
        const float an = e * sPhi[n];
        sW[idx * 2]     = rr * cosf(an);
        sW[idx * 2 + 1] = rr * sinf(an);
    }
    for (int idx = tid; idx < CHUNK * NPOLE; idx += 256) {   // E table
        const int t = idx >> 4, n = idx & 15;
        const float e  = (float)(t + 1);
        const float rr = exp2f(e * sL2r[n]);
        const float pr = rr * cosf(e * sPhi[n]);
        const float pi = rr * sinf(e * sPhi[n]);
        sEE[t * 32 + n]      =  sGr[n] * pr - sGi[n] * pi;   //  Re(gp q^{t+1})
        sEE[t * 32 + 16 + n] = -(sGr[n] * pi + sGi[n] * pr); // -Im(gp q^{t+1})
    }
    if (tid < 128) {                                         // reversed, zero-padded taps
        if (tid < 64) {
            const int t = tid;                               // tau
            float acc = 0.f;
            #pragma unroll
            for (int n = 0; n < NPOLE; ++n) {
                const float rr = exp2f((float)t * sL2r[n]);
                const float an = (float)t * sPhi[n];
                acc += sGr[n] * rr * cosf(an) - sGi[n] * rr * sinf(an);
            }
            sTapR[63 - t] = acc;                             // tapR[j] = k[63-j]
        } else {
            sTapR[tid] = 0.f;                                // strict-upper zeros
        }
    }
    __syncthreads();

    // ---- stage 3: chunk-level scan (serial only over 32 chunks) ----
    // wave wv handles batch b=wv; x chunks staged LDS-side with async loads, one chunk ahead.
    if (wv < Bn) {
        const float* px = x + ((size_t)wv * (size_t)D + (size_t)d) * (size_t)SEQL;
        const uint32_t xb[2] = { (uint32_t)(uintptr_t)&sXA[wv][0][0],
                                 (uint32_t)(uintptr_t)&sXA[wv][1][0] };
        // prologue: fetch chunk 0
        async_ld_b64(xb[0] + (uint32_t)(lane * 8), px + lane * 2);

        float Sr = 0.f, Si = 0.f;
        for (int c = 0; c < NCH; ++c) {
            if (lane < 16) {                       // store state BEFORE chunk c
                float* ss = &sSS[(wv * NCH + c) * 32];
                ss[hn] = Sr; ss[16 + hn] = Si;
            }
            if (c + 1 < NCH) {                     // prefetch next chunk, then wait for current
                async_ld_b64(xb[(c + 1) & 1] + (uint32_t)(lane * 8),
                             px + (c + 1) * CHUNK + lane * 2);
                asm volatile("s_wait_asynccnt 0x1" ::: "memory");
            } else {
                asm volatile("s_wait_asynccnt 0x0" ::: "memory");
            }
            float ar = 0.f, ai = 0.f;
            const float* xc = &sXA[wv][c & 1][half * 32];
            const float* wp = &sW[(hn * CHUNK + half * 32) * 2];
            #pragma unroll 8
            for (int uu = 0; uu < 32; ++uu) {
                const float xv = xc[uu];
                ar = fmaf(wp[uu * 2],     xv, ar);
                ai = fmaf(wp[uu * 2 + 1], xv, ai);
            }
            ar += __shfl_xor(ar, 16, 32);          // combine the two halves
            ai += __shfl_xor(ai, 16, 32);
            const float nr = sQCr[hn] * Sr - sQCi[hn] * Si + ar;
            const float ni = sQCr[hn] * Si + sQCi[hn] * Sr + ai;
            Sr = nr; Si = ni;
        }
    }
    __syncthreads();

    // ---- stage 4: WMMA pass ----
    // y(64 x B*32 cols) = [Toeplitz(tap) 64x64 | E 64x32] x [x_chunk ; S], fp32 16x16x4 steps.
    const float om = omega[d];
    for (int i = 0; i < Bn; ++i) {
        const int tile = wv * Bn + i;       // 4 row-tiles x (2*Bn) col-tiles
        const int m   = tile & 3;           // row tile (t block)
        const int j   = tile >> 2;          // column tile
        const int col = 16 * j + hn;        // global column = b*32 + c
        const int b2  = col >> 5;
        const int cc  = col & 31;
        const int t   = 16 * m + hn;        // A-fragment row for this lane

        const float* pxc  = x + ((size_t)b2 * (size_t)D + (size_t)d) * (size_t)SEQL + cc * CHUNK;
        const float* ssc  = &sSS[(b2 * NCH + cc) * 32];
        const float* tapb = &sTapR[63 - t + 2 * half];  // unconditional Toeplitz base
        const float* eeb  = &sEE[t * 32 + 2 * half];

        v8f acc = {0.f, 0.f, 0.f, 0.f, 0.f, 0.f, 0.f, 0.f};
        #pragma unroll
        for (int k0 = 0; k0 < 96; k0 += 4) {
            float a0, a1, b0, b1;
            if (k0 < 64) {                                   // Toeplitz block (padded table, no branches)
                a0 = tapb[k0];
                a1 = tapb[k0 + 1];
                const float2 v = *(const float2*)(pxc + k0 + 2 * half);
                b0 = v.x; b1 = v.y;
            } else {                                         // E block x state rows [Sre|Sim]
                a0 = eeb[k0 - 64];
                a1 = eeb[k0 - 63];
                b0 = ssc[k0 - 64 + 2 * half];
                b1 = ssc[k0 - 63 + 2 * half];
            }
            v2f av = {a0, a1};
            v2f bv = {b0, b1};
            acc = __builtin_amdgcn_wmma_f32_16x16x4_f32(false, av, false, bv,
                                                        (short)0, acc, false, false);
        }

        // C/D layout: lane<16 -> rows m*16+0..7, lane>=16 -> rows m*16+8..15; N = hn.
        const size_t obase = ((size_t)b2 * (size_t)D + (size_t)d) * (size_t)SEQL
                           + (size_t)(cc * CHUNK + 16 * m + 8 * half);
        const float4 xa = *(const float4*)(x + obase);
        const float4 xb4 = *(const float4*)(x + obase + 4);
        float4 o0, o1;
        o0.x = acc[0] + om * xa.x;  o0.y = acc[1] + om * xa.y;
        o0.z = acc[2] + om * xa.z;  o0.w = acc[3] + om * xa.w;
        o1.x = acc[4] + om * xb4.x; o1.y = acc[5] + om * xb4.y;
        o1.z = acc[6] + om * xb4.z; o1.w = acc[7] + om * xb4.w;
        *(float4*)(out + obase)     = o0;
        *(float4*)(out + obase + 4) = o1;
    }
}

extern "C" void kernel_launch(void* const* d_in, const int* in_sizes, int n_in,
                              void* d_out, int out_size, void* d_ws, size_t ws_size,
                              hipStream_t stream) {
    const float* x     = (const float*)d_in[0];
    const float* alpha = (const float*)d_in[1];
    const float* delta = (const float*)d_in[2];
    const float* theta = (const float*)d_in[3];
    const float* gr    = (const float*)d_in[4];
    const float* gi    = (const float*)d_in[5];
    const float* omega = (const float*)d_in[6];
    float* out = (float*)d_out;

    const int D  = in_sizes[3];                       // theta has D elements
    int Bn = in_sizes[0] / (D * SEQL);                // x = B*D*L
    if (Bn > 8) Bn = 8;                               // LDS state block sized for B<=8

    dim3 grid(D), block(256);
    hipLaunchKernelGGL(cema_scan_wmma_kernel, grid, block, 0, stream,
                       x, alpha, delta, theta, gr, gi, omega, out, Bn, D);
}